// LIF_33225867001926
// MI455X (gfx1250) — compile-verified
//
#include <hip/hip_runtime.h>

// LIF forward: spikes[t] = (mem_t > 0.5) ? 1 : 0, mem_{t+1} = mem_t/2 + x - spike
// Pure streaming recurrence: bandwidth-bound (~103 MB @ 23.3 TB/s => ~4.4us floor).
// No matrix structure -> WMMA inapplicable; use CDNA5 async global->LDS copy
// engine (ASYNCcnt) for the load path + 128-bit NT stores.

typedef float v4f __attribute__((ext_vector_type(4)));
typedef int   v4i_vs __attribute__((vector_size(16)));   // matches builtin param type

typedef __attribute__((address_space(1))) v4i_vs* gas_v4i_p;  // global (AS1)
typedef __attribute__((address_space(3))) v4i_vs* lds_v4i_p;  // LDS (AS3)

#define LIF_T      4
#define LIF_BLOCK  256

// CDNA5 async global->LDS path (gfx1250 only), guarded so we always compile.
#if defined(__gfx1250__) && __has_builtin(__builtin_amdgcn_global_load_async_to_lds_b128)
#define LIF_ASYNC 1
#else
#define LIF_ASYNC 0
#endif

__device__ __forceinline__ void lif_wait_async() {
#if defined(__gfx1250__)
#if __has_builtin(__builtin_amdgcn_s_wait_asynccnt)
  __builtin_amdgcn_s_wait_asynccnt(0);
#else
  asm volatile("s_wait_asynccnt 0" ::: "memory");
#endif
#endif
}

__device__ __forceinline__ v4f lif_spike4(v4f m) {
  v4f s;
  s.x = (m.x > 0.5f) ? 1.0f : 0.0f;
  s.y = (m.y > 0.5f) ? 1.0f : 0.0f;
  s.z = (m.z > 0.5f) ? 1.0f : 0.0f;
  s.w = (m.w > 0.5f) ? 1.0f : 0.0f;
  return s;
}

__global__ __launch_bounds__(LIF_BLOCK)
void lif_fwd_v4(const float* __restrict__ x, float* __restrict__ out,
                long long n4, long long stride) {
  const long long i = (long long)blockIdx.x * LIF_BLOCK + threadIdx.x;
  if (i >= n4) return;
  const long long base = i * 4;

  v4f xv[LIF_T];

#if LIF_ASYNC
  // Stage all T tiles through LDS with the async DMA engine; each lane copies
  // and later reads only its own 16B, so a single s_wait_asynccnt suffices
  // (no workgroup barrier needed).
  __shared__ v4f tile[LIF_T][LIF_BLOCK];
#pragma unroll
  for (int t = 0; t < LIF_T; ++t) {
    float* g = const_cast<float*>(x + (long long)t * stride + base);
    __builtin_amdgcn_global_load_async_to_lds_b128(
        (gas_v4i_p)g,
        (lds_v4i_p)&tile[t][threadIdx.x],
        0, 0);
  }
  lif_wait_async();
#pragma unroll
  for (int t = 0; t < LIF_T; ++t) {
    xv[t] = tile[t][threadIdx.x];   // ds_load_b128
  }
#else
  // Fallback: direct 128-bit nontemporal loads (single-pass stream).
#pragma unroll
  for (int t = 0; t < LIF_T; ++t) {
    xv[t] = __builtin_nontemporal_load(
        (const v4f*)(x + (long long)t * stride + base));
  }
#endif

  v4f mem = {0.0f, 0.0f, 0.0f, 0.0f};
#pragma unroll
  for (int t = 0; t < LIF_T; ++t) {
    mem = mem * 0.5f + xv[t];          // mem/TAU + x_t  (TAU=2, exact)
    v4f s = lif_spike4(mem);           // round(clip(mem,0,1)) with VTH=1
    mem = mem - s;                     // detach_reset (forward identical)
    __builtin_nontemporal_store(
        s, (v4f*)(out + (long long)t * stride + base));
  }
}

// Scalar tail for stride % 4 != 0 (not hit for the reference shape, kept for
// generality; deterministic).
__global__ void lif_fwd_tail(const float* __restrict__ x, float* __restrict__ out,
                             long long start, long long stride) {
  const long long i = start + (long long)blockIdx.x * blockDim.x + threadIdx.x;
  if (i >= stride) return;
  float mem = 0.0f;
#pragma unroll
  for (int t = 0; t < LIF_T; ++t) {
    mem = mem * 0.5f + x[(long long)t * stride + i];
    const float s = (mem > 0.5f) ? 1.0f : 0.0f;
    mem -= s;
    out[(long long)t * stride + i] = s;
  }
}

extern "C" void kernel_launch(void* const* d_in, const int* in_sizes, int n_in,
                              void* d_out, int out_size, void* d_ws, size_t ws_size,
                              hipStream_t stream) {
  (void)n_in; (void)d_ws; (void)ws_size; (void)out_size;

  const float* x = (const float*)d_in[0];
  float* out = (float*)d_out;

  const long long total  = (long long)in_sizes[0]; // T * B*C*H*W
  const long long stride = total / LIF_T;          // elements per timestep
  const long long n4     = stride / 4;             // float4 columns
  const long long rem    = stride - n4 * 4;

  if (n4 > 0) {
    const long long blocks = (n4 + LIF_BLOCK - 1) / LIF_BLOCK;
    lif_fwd_v4<<<(unsigned int)blocks, LIF_BLOCK, 0, stream>>>(x, out, n4, stride);
  }
  if (rem > 0) {
    lif_fwd_tail<<<1, 64, 0, stream>>>(x, out, n4 * 4, stride);
  }
}